// GraphGRUCell_68676527063546
// MI455X (gfx1250) — compile-verified
//
#include <hip/hip_runtime.h>
#include <hip/hip_bf16.h>
#include <math.h>

typedef float v2f __attribute__((ext_vector_type(2)));
typedef float v8f __attribute__((ext_vector_type(8)));

namespace {
constexpr int   NN        = 50000;   // nodes
constexpr int   EE        = 800000;  // edges
constexpr int   HF        = 128;     // H*F
constexpr float NEG_SLOPE = 0.2f;
constexpr int   NGATE     = 2;       // only u,c gates matter (r is dead code)
}

static_assert(NN % 16 == 0, "M tiles exact");

__device__ __forceinline__ float lrelu(float v) {
  return v >= 0.0f ? v : NEG_SLOPE * v;
}
__device__ __forceinline__ float sigmoidf_(float v) {
  return 1.0f / (1.0f + __expf(-v));
}
// classic signed-max / unsigned-min trick: works with -inf init, mixed signs
__device__ __forceinline__ void atomicMaxF(float* addr, float value) {
  if (__float_as_int(value) >= 0) {
    atomicMax((int*)addr, __float_as_int(value));
  } else {
    atomicMin((unsigned int*)addr, __float_as_uint(value));
  }
}
__device__ __forceinline__ void atomAddF(float* addr, float v) {
  unsafeAtomicAdd(addr, v);   // lowers to global_atomic_add_f32 on CDNA
}

// ---------------------------------------------------------------------------
// 0) init: zero accumulators, -inf segment-max, zero segment-sum
// ---------------------------------------------------------------------------
__global__ void init_kernel(float* __restrict__ acc, float* __restrict__ mbuf,
                            float* __restrict__ sbuf) {
  int idx = blockIdx.x * blockDim.x + threadIdx.x;
  if (idx < NGATE * NN * HF) acc[idx] = 0.0f;
  if (idx < NGATE * NN * 2) {
    mbuf[idx] = -INFINITY;
    sbuf[idx] = 0.0f;
  }
}

// ---------------------------------------------------------------------------
// 1) z[g] = x @ W[g+1]^T  via V_WMMA_F32_16X16X4_F32
//    block = 256 threads = 8 waves; each wave owns one 16-col tile of the
//    128 outputs; block owns 16 rows of x staged through LDS.
// ---------------------------------------------------------------------------
__global__ __launch_bounds__(256) void gemm_wmma_kernel(
    const float* __restrict__ x, const float* __restrict__ W,
    float* __restrict__ z) {
  const int rowTile = blockIdx.x;        // 0..N/16-1
  const int g       = blockIdx.y;        // 0 -> u (W[1]), 1 -> c (W[2])
  const float* Wg = W + (size_t)(g + 1) * 128 * 128;
  float* zg       = z + (size_t)g * NN * HF;

  __shared__ float As[16][132];          // pad 132: kills 16-way bank conflicts
  const int tid = threadIdx.x;
  const float* xrow = x + (size_t)rowTile * 16 * 128;
  for (int i = tid; i < 16 * 128; i += 256)
    As[i >> 7][i & 127] = xrow[i];
  __syncthreads();

  const int wave    = tid >> 5;          // 0..7 column tile
  const int lane    = tid & 31;
  const int colBase = wave * 16;
  const int mrow    = lane & 15;         // M (for A) / N (for B) index
  const int khalf   = (lane >> 4) * 2;   // lanes 0-15 -> K+0/1, 16-31 -> K+2/3

  v8f acc = {};
  for (int k = 0; k < 128; k += 4) {
    v2f a, b;
    // A 16x4 fp32 fragment (ISA 7.12.2): v0 = K{0|2}, v1 = K{1|3}
    a.x = As[mrow][k + khalf];
    a.y = As[mrow][k + khalf + 1];
    // B[k'][n] = W[n][k'] (z = x @ W^T); same half-lane K striping
    const int n = colBase + mrow;
    b.x = Wg[(size_t)n * 128 + k + khalf];
    b.y = Wg[(size_t)n * 128 + k + khalf + 1];
    acc = __builtin_amdgcn_wmma_f32_16x16x4_f32(
        /*neg_a=*/false, a, /*neg_b=*/false, b,
        /*c_mod=*/(short)0, acc, /*reuse_a=*/false, /*reuse_b=*/false);
  }

  // D 16x16 f32: VGPR r -> M=r (lanes 0-15) / M=8+r (lanes 16-31), N=lane%16
  const int nOut  = colBase + mrow;
  const int mBase = (lane >> 4) * 8;
  float* zt = zg + (size_t)rowTile * 16 * 128;
#pragma unroll
  for (int r = 0; r < 8; ++r)
    zt[(size_t)(mBase + r) * 128 + nOut] = acc[r];
}

// ---------------------------------------------------------------------------
// 2) el/er: per (gate, node, head) 64-wide dot with attn vectors
// ---------------------------------------------------------------------------
__global__ void attn_kernel(const float* __restrict__ z,
                            const float* __restrict__ attn_l,
                            const float* __restrict__ attn_r,
                            float* __restrict__ el, float* __restrict__ er) {
  int idx = blockIdx.x * blockDim.x + threadIdx.x;  // over NGATE*N*2
  if (idx >= NGATE * NN * 2) return;
  const int h = idx & 1;
  const int n = (idx >> 1) % NN;
  const int g = idx / (NN * 2);
  const float* zr = z + ((size_t)g * NN + n) * HF + h * 64;
  const float* al = attn_l + (size_t)(g + 1) * HF + h * 64;
  const float* ar = attn_r + (size_t)(g + 1) * HF + h * 64;
  float sl = 0.0f, sr = 0.0f;
#pragma unroll 8
  for (int f = 0; f < 64; ++f) {
    const float zv = zr[f];
    sl = fmaf(zv, al[f], sl);
    sr = fmaf(zv, ar[f], sr);
  }
  el[idx] = sl;
  er[idx] = sr;
}

// ---------------------------------------------------------------------------
// 3) segment-max of edge logits over dst
// ---------------------------------------------------------------------------
__global__ void edge_max_kernel(const int* __restrict__ src,
                                const int* __restrict__ dst,
                                const float* __restrict__ el,
                                const float* __restrict__ er,
                                float* __restrict__ mbuf) {
  int idx = blockIdx.x * blockDim.x + threadIdx.x;  // over NGATE*E*2
  if (idx >= NGATE * EE * 2) return;
  const int h = idx & 1;
  const int e = (idx >> 1) % EE;
  const int g = idx / (EE * 2);
  const int s = src[e], d = dst[e];
  const float v = lrelu(el[((size_t)g * NN + s) * 2 + h] +
                        er[((size_t)g * NN + d) * 2 + h]);
  atomicMaxF(&mbuf[((size_t)g * NN + d) * 2 + h], v);
}

// ---------------------------------------------------------------------------
// 4) segment-sum of exp(logit - max) over dst
// ---------------------------------------------------------------------------
__global__ void edge_sum_kernel(const int* __restrict__ src,
                                const int* __restrict__ dst,
                                const float* __restrict__ el,
                                const float* __restrict__ er,
                                const float* __restrict__ mbuf,
                                float* __restrict__ sbuf) {
  int idx = blockIdx.x * blockDim.x + threadIdx.x;
  if (idx >= NGATE * EE * 2) return;
  const int h = idx & 1;
  const int e = (idx >> 1) % EE;
  const int g = idx / (EE * 2);
  const int s = src[e], d = dst[e];
  const size_t nd = ((size_t)g * NN + d) * 2 + h;
  const float v = lrelu(el[((size_t)g * NN + s) * 2 + h] + er[nd]);
  atomAddF(&sbuf[nd], __expf(v - mbuf[nd]));
}

// ---------------------------------------------------------------------------
// 5) weighted message scatter: one wave32 per (gate, edge);
//    lane -> 4 contiguous feats (float4 gather, 4x f32 atomics)
// ---------------------------------------------------------------------------
__global__ __launch_bounds__(256) void edge_scatter_kernel(
    const int* __restrict__ src, const int* __restrict__ dst,
    const float* __restrict__ el, const float* __restrict__ er,
    const float* __restrict__ mbuf, const float* __restrict__ sbuf,
    const float* __restrict__ z, float* __restrict__ acc) {
  const long long gtid = (long long)blockIdx.x * 256 + threadIdx.x;
  const long long widx = gtid >> 5;              // wave index over NGATE*E
  if (widx >= (long long)NGATE * EE) return;
  const int lane = threadIdx.x & 31;
  const int e = (int)(widx % EE);
  const int g = (int)(widx / EE);
  const int s = src[e], d = dst[e];
  const int fbase = lane * 4;                    // 0..124
  const int h = fbase >> 6;                      // lanes 0-15: h=0, 16-31: h=1
  const size_t nd = ((size_t)g * NN + d) * 2 + h;
  float logit = lrelu(el[((size_t)g * NN + s) * 2 + h] + er[nd]);
  const float alpha = __expf(logit - mbuf[nd]) / sbuf[nd];
  const float4 zv =
      *(const float4*)(z + ((size_t)g * NN + s) * HF + fbase);
  float* ap = acc + ((size_t)g * NN + d) * HF + fbase;
  atomAddF(ap + 0, alpha * zv.x);
  atomAddF(ap + 1, alpha * zv.y);
  atomAddF(ap + 2, alpha * zv.z);
  atomAddF(ap + 3, alpha * zv.w);
}

// ---------------------------------------------------------------------------
// 6) finalize: u = sigmoid(acc_u + biases), c = sigmoid(acc_c + biases),
//    new_h = u*h + (1-u)*c
// ---------------------------------------------------------------------------
__global__ void finalize_kernel(const float* __restrict__ acc,
                                const float* __restrict__ h_in,
                                const float* __restrict__ conv_bias,
                                const float* __restrict__ gate_bias,
                                float* __restrict__ out) {
  int idx = blockIdx.x * blockDim.x + threadIdx.x;  // over N*128
  if (idx >= NN * HF) return;
  const int f = idx & 127;
  const float uval =
      sigmoidf_(acc[idx] + conv_bias[128 + f] + gate_bias[128 + f]);
  const float cval = sigmoidf_(acc[(size_t)NN * HF + idx] +
                               conv_bias[256 + f] + gate_bias[256 + f]);
  const float hv = h_in[idx];
  out[idx] = uval * hv + (1.0f - uval) * cval;
}

// ---------------------------------------------------------------------------
extern "C" void kernel_launch(void* const* d_in, const int* in_sizes, int n_in,
                              void* d_out, int out_size, void* d_ws,
                              size_t ws_size, hipStream_t stream) {
  (void)in_sizes; (void)n_in; (void)out_size; (void)ws_size;
  const float* x         = (const float*)d_in[0];
  const float* h         = (const float*)d_in[1];
  const float* W         = (const float*)d_in[2];
  const float* attn_l    = (const float*)d_in[3];
  const float* attn_r    = (const float*)d_in[4];
  const float* conv_bias = (const float*)d_in[5];
  const float* gate_bias = (const float*)d_in[6];
  const int*   src       = (const int*)d_in[7];
  const int*   dst       = (const int*)d_in[8];
  float* out = (float*)d_out;

  // workspace layout (floats): z[2][N][128] | acc[2][N][128] | el/er/m/s[2][N][2]
  float* ws   = (float*)d_ws;
  float* z    = ws;
  float* acc  = z    + (size_t)NGATE * NN * HF;
  float* el   = acc  + (size_t)NGATE * NN * HF;
  float* er   = el   + (size_t)NGATE * NN * 2;
  float* mbuf = er   + (size_t)NGATE * NN * 2;
  float* sbuf = mbuf + (size_t)NGATE * NN * 2;

  // 0) init
  {
    const int n = NGATE * NN * HF;
    init_kernel<<<(n + 255) / 256, 256, 0, stream>>>(acc, mbuf, sbuf);
  }
  // 1) WMMA projections (u, c gates)
  {
    dim3 grid(NN / 16, NGATE);
    gemm_wmma_kernel<<<grid, 256, 0, stream>>>(x, W, z);
  }
  // 2) attention scalars
  {
    const int n = NGATE * NN * 2;
    attn_kernel<<<(n + 255) / 256, 256, 0, stream>>>(z, attn_l, attn_r, el, er);
  }
  // 3) edge softmax max
  {
    const int n = NGATE * EE * 2;
    edge_max_kernel<<<(n + 255) / 256, 256, 0, stream>>>(src, dst, el, er, mbuf);
  }
  // 4) edge softmax denom
  {
    const int n = NGATE * EE * 2;
    edge_sum_kernel<<<(n + 255) / 256, 256, 0, stream>>>(src, dst, el, er,
                                                         mbuf, sbuf);
  }
  // 5) message scatter (wave per gate-edge)
  {
    const long long waves = (long long)NGATE * EE;
    const long long threads = waves * 32;
    const int blocks = (int)((threads + 255) / 256);
    edge_scatter_kernel<<<blocks, 256, 0, stream>>>(src, dst, el, er, mbuf,
                                                    sbuf, z, acc);
  }
  // 6) GRU combine
  {
    const int n = NN * HF;
    finalize_kernel<<<(n + 255) / 256, 256, 0, stream>>>(acc, h, conv_bias,
                                                         gate_bias, out);
  }
}